// ToDense_35931696398508
// MI455X (gfx1250) — compile-verified
//
#include <hip/hip_runtime.h>

// ---- Problem geometry (mirrors the reference module config) ----
#define DIM_B 2
#define DIM_C 128
#define DIM_X 11
#define DIM_Y 200
#define DIM_Z 176
#define SPATIAL (DIM_X * DIM_Y * DIM_Z)   // 387200

typedef float  float4v __attribute__((ext_vector_type(4)));
typedef int    int4v   __attribute__((ext_vector_type(4)));

// -----------------------------------------------------------------------------
// Kernel 1: streaming zero-fill of the dense output (396 MB).
// 32-bit indexing (out_size = 99.1M < 2^31) keeps the hot path to a handful of
// VALU ops + one global_store_b128 th:TH_STORE_NT. NT keeps the 192 MB L2 free
// for write-combining the scatter's partial-line writes that follow.
// -----------------------------------------------------------------------------
__global__ void todense_zero_kernel(float* __restrict__ out, unsigned n) {
    unsigned i = (blockIdx.x * blockDim.x + threadIdx.x) * 4u;
    if (i + 4u <= n) {
        float4v z = {0.0f, 0.0f, 0.0f, 0.0f};
        __builtin_nontemporal_store(z, (float4v*)(out + i));
    } else if (i < n) {
        // Generic tail (not taken for this config: n % 4 == 0).
        for (unsigned j = i; j < n; ++j) out[j] = 0.0f;
    }
}

// -----------------------------------------------------------------------------
// Kernel 2: scatter. One wave32 per point:
//   - the (b,x,y,z) quad is fetched once per wave via the scalar path
//     (readfirstlane -> uniform -> s_load_b128, SALU decode, s_cbranch OOB skip)
//   - lanes cooperatively load the point's 128 channels as 32 x float4
//     (one coalesced 512B global_load_b128 per wave, non-temporal: read-once)
//   - each lane stores 4 dwords at channel stride SPATIAL from ONE base
//     address using compile-time immediate offsets (1.55/3.1/4.65 MB, all
//     within the signed 24-bit IOFFSET field). Default RT hint so L2
//     write-combines the ~8 points that dirty each 128B z-line.
// -----------------------------------------------------------------------------
__global__ void todense_scatter_kernel(const float* __restrict__ feat,
                                       const int*   __restrict__ idx,
                                       float*       __restrict__ out,
                                       int n_points) {
    const int lane  = threadIdx.x & 31;          // wave32 (gfx1250)
    const int wave  = threadIdx.x >> 5;
    const int point = blockIdx.x * (blockDim.x >> 5) + wave;
    if (point >= n_points) return;

    // 'point' is wave-uniform by construction; make that explicit so the
    // backend can use the scalar memory path for the index quad.
    const int upoint = __builtin_amdgcn_readfirstlane(point);
    int4v q = *((const int4v*)idx + upoint);
    const int b = q.x, x = q.y, y = q.z, z = q.w;

    // Reference semantics: out-of-bounds x/y/z points are dropped (uniform
    // branch -> whole wave exits via s_cbranch).
    if ((unsigned)x >= (unsigned)DIM_X ||
        (unsigned)y >= (unsigned)DIM_Y ||
        (unsigned)z >= (unsigned)DIM_Z) return;

    // base = b*C*S + spatial ; max flat index 99.1M < 2^31 -> 32-bit math.
    const int spatial = x * (DIM_Y * DIM_Z) + y * DIM_Z + z;
    const int base    = b * (DIM_C * SPATIAL) + spatial;

    // Coalesced, read-once feature fetch: lane owns channels 4*lane..4*lane+3.
    const float4v* f4 = (const float4v*)(feat + (long long)upoint * DIM_C);
    float4v v = __builtin_nontemporal_load(f4 + lane);

    // One base pointer per lane; the three +SPATIAL offsets become instruction
    // immediate offsets on global_store_b32.
    float* p = out + base + (lane * 4) * SPATIAL;
    p[0 * SPATIAL] = v.x;
    p[1 * SPATIAL] = v.y;
    p[2 * SPATIAL] = v.z;
    p[3 * SPATIAL] = v.w;
}

// -----------------------------------------------------------------------------
// Launch: zero-fill then scatter, ordered on the same stream.
// -----------------------------------------------------------------------------
extern "C" void kernel_launch(void* const* d_in, const int* in_sizes, int n_in,
                              void* d_out, int out_size, void* d_ws, size_t ws_size,
                              hipStream_t stream) {
    const float* feat = (const float*)d_in[0];   // [N, 128] f32
    const int*   idx  = (const int*)d_in[1];     // [N, 4]   i32 (b,x,y,z)
    float*       out  = (float*)d_out;           // [B, C, X, Y, Z] f32

    const int n_points = in_sizes[1] / 4;

    // Zero-fill: one float4 per thread.
    {
        unsigned nelem = (unsigned)out_size;
        unsigned nvec4 = (nelem + 3u) / 4u;
        int threads = 256;
        unsigned blocks = (nvec4 + threads - 1) / threads;
        todense_zero_kernel<<<blocks, threads, 0, stream>>>(out, nelem);
    }

    // Scatter: 8 waves per 256-thread block -> 8 points per block.
    {
        int threads = 256;
        int waves_per_block = threads / 32;
        int blocks = (n_points + waves_per_block - 1) / waves_per_block;
        todense_scatter_kernel<<<blocks, threads, 0, stream>>>(feat, idx, out, n_points);
    }
}